// GINEConv_87806311399694
// MI455X (gfx1250) — compile-verified
//
#include <hip/hip_runtime.h>

#define N_NODES 100000
#define N_EDGES 640000
#define D 128

typedef __attribute__((ext_vector_type(2))) float v2f;
typedef __attribute__((ext_vector_type(8))) float v8f;

// ---------------------------------------------------------------------------
// K1: h0 = (1 + eps) * x        (12.8M floats, exact cover: 12500 * 256 * 4)
// ---------------------------------------------------------------------------
__global__ __launch_bounds__(256) void init_h0(const float* __restrict__ x,
                                               const float* __restrict__ eps_p,
                                               float* __restrict__ h0) {
    const float s = 1.0f + eps_p[0];
    const size_t i = ((size_t)blockIdx.x * blockDim.x + threadIdx.x) * 4;
    float4 v = *(const float4*)(x + i);
    v.x *= s; v.y *= s; v.z *= s; v.w *= s;
    *(float4*)(h0 + i) = v;
}

// ---------------------------------------------------------------------------
// K2: per-edge message + scatter-add.
// One wave (32 lanes) per edge; each lane handles 4 consecutive floats.
// msg = relu(x[src] + edge_attr[e]); h0[dst] += msg  (hw fp32 atomics)
// 80000 blocks * 8 waves = 640000 edges exactly.
// ---------------------------------------------------------------------------
__global__ __launch_bounds__(256) void edge_scatter(const float* __restrict__ x,
                                                    const int* __restrict__ ei,
                                                    const float* __restrict__ ea,
                                                    float* __restrict__ h0) {
    const int wave = threadIdx.x >> 5;
    const int lane = threadIdx.x & 31;
    const int e    = blockIdx.x * 8 + wave;           // < 640000 exactly

    const int src = ei[e];
    const int dst = ei[N_EDGES + e];

    const size_t c = (size_t)lane * 4;
    float4 xv = *(const float4*)(x  + (size_t)src * D + c);
    float4 av = *(const float4*)(ea + (size_t)e   * D + c);

    float4 m;
    m.x = fmaxf(xv.x + av.x, 0.0f);
    m.y = fmaxf(xv.y + av.y, 0.0f);
    m.z = fmaxf(xv.z + av.z, 0.0f);
    m.w = fmaxf(xv.w + av.w, 0.0f);

    float* o = h0 + (size_t)dst * D + c;
    unsafeAtomicAdd(o + 0, m.x);      // -> global_atomic_add_f32
    unsafeAtomicAdd(o + 1, m.y);
    unsafeAtomicAdd(o + 2, m.z);
    unsafeAtomicAdd(o + 3, m.w);
}

// ---------------------------------------------------------------------------
// K3/K4: out[M,128] = act(A[M,128] @ W[128,128] + bias)  via fp32 WMMA.
//
// Persistent blocks: 625 blocks x TILES(=10) 16-row tiles = 6250 tiles.
// Block = 256 threads (8 waves); wave w owns output columns [16w, 16w+16).
//
// Per block:  stage W[128][128] -> LDS once, pull this wave's B column-slab
// into 32 v2f register fragments (64 VGPRs), then loop tiles:
//   stage A tile (double-buffered LDS), one barrier, then 32 x
//   { ds_load_b64 (A pair) ; v_wmma_f32_16x16x4_f32 } — minimal DS per WMMA.
//
// f32 WMMA fragment layouts (ISA 7.12.2):
//   A (16x4):  lane l, vgpr j  ->  A[l%16][2*(l>>4)+j]
//   B (4x16):  lane l, vgpr j  ->  B[2*(l>>4)+j][l%16]
//   C/D:       lane l, vgpr r  ->  C[r + 8*(l>>4)][l%16]
//
// LDS strides: W rows 136 (2*136 % 64 == 16 -> lane-halves hit disjoint bank
// groups), A rows 132 (132 % 64 == 4 -> 16 lanes hit distinct banks).
// LDS total: (128*136 + 2*16*132)*4 = 86,528 B  (WGP allows 320 KB).
// ---------------------------------------------------------------------------
#define SW 136          // LDS row stride for W
#define SA 132          // LDS row stride for A tile
#define TILES 10        // 16-row tiles per block
#define GEMM_BLOCKS 625 // 625 * 10 * 16 = 100000 rows exactly

__global__ __launch_bounds__(256) void gemm128(const float* __restrict__ A,
                                               const float* __restrict__ W,
                                               const float* __restrict__ bias,
                                               float* __restrict__ out,
                                               int do_relu) {
    extern __shared__ float smem[];
    float* lw  = smem;                 // 128 * SW floats
    float* la0 = smem + 128 * SW;      // 16 * SA floats
    float* la1 = la0 + 16 * SA;        // 16 * SA floats

    const int t = threadIdx.x;

    // Stage W[128][128] -> lw (float4, padded rows; SW*4B is 16B-aligned)
    for (int i4 = t * 4; i4 < 128 * 128; i4 += 256 * 4) {
        const int r = i4 >> 7, c = i4 & 127;
        *(float4*)(lw + r * SW + c) = *(const float4*)(W + i4);
    }
    __syncthreads();

    const int wave = t >> 5;        // 0..7  -> N tile
    const int lane = t & 31;
    const int half = lane >> 4;     // 0/1   -> K sub-pair
    const int lm   = lane & 15;     // M for A, N for B/C/D
    const int n0   = wave * 16;

    // Pull this wave's B column-slab into registers: 32 x v2f = 64 VGPRs.
    v2f bfrag[32];
#pragma unroll
    for (int kk = 0; kk < 32; ++kk) {
        const int k = kk * 4 + half * 2;
        bfrag[kk].x = lw[k * SW + n0 + lm];
        bfrag[kk].y = lw[(k + 1) * SW + n0 + lm];
    }
    const float bv = bias[n0 + lm];

    for (int ti = 0; ti < TILES; ++ti) {
        float* la = (ti & 1) ? la1 : la0;
        const size_t row0 = ((size_t)blockIdx.x * TILES + ti) * 16;

        // Stage A tile [16][128] -> la (2 float4 per thread, coalesced).
        {
            int i4 = t * 4;
            int r = i4 >> 7, c = i4 & 127;
            *(float4*)(la + r * SA + c) = *(const float4*)(A + row0 * D + i4);
            i4 += 1024; r = i4 >> 7; c = i4 & 127;
            *(float4*)(la + r * SA + c) = *(const float4*)(A + row0 * D + i4);
        }
        // One barrier per tile: previous tile's compute (other buffer) is
        // complete for every wave that reaches this point.
        __syncthreads();

        v8f acc = {};
#pragma unroll
        for (int kk = 0; kk < 32; ++kk) {
            const int k = kk * 4 + half * 2;
            v2f a;
            a.x = la[lm * SA + k];      // ds_load_b64 (8B aligned: SA*4, k even)
            a.y = la[lm * SA + k + 1];
            acc = __builtin_amdgcn_wmma_f32_16x16x4_f32(false, a, false, bfrag[kk],
                                                        (short)0, acc, false, false);
        }

#pragma unroll
        for (int r = 0; r < 8; ++r) {
            const int m = r + 8 * half;
            float v = acc[r] + bv;
            if (do_relu) v = fmaxf(v, 0.0f);
            out[(row0 + m) * D + n0 + lm] = v;
        }
    }
}

// ---------------------------------------------------------------------------
extern "C" void kernel_launch(void* const* d_in, const int* in_sizes, int n_in,
                              void* d_out, int out_size, void* d_ws, size_t ws_size,
                              hipStream_t stream) {
    const float* x   = (const float*)d_in[0];
    const int*   ei  = (const int*)  d_in[1];
    const float* ea  = (const float*)d_in[2];
    const float* eps = (const float*)d_in[3];
    const float* W1  = (const float*)d_in[4];
    const float* b1  = (const float*)d_in[5];
    const float* W2  = (const float*)d_in[6];
    const float* b2  = (const float*)d_in[7];
    float* out = (float*)d_out;

    float* h0 = (float*)d_ws;                    // [N_NODES, D]
    float* h1 = h0 + (size_t)N_NODES * D;        // [N_NODES, D]

    const size_t lds_bytes = (size_t)(128 * SW + 2 * 16 * SA) * sizeof(float); // 86,528 B

    // K1: h0 = (1+eps)*x            12,800,000 floats = 12500*256*4 exactly
    init_h0<<<12500, 256, 0, stream>>>(x, eps, h0);

    // K2: scatter-add messages      640000 edges = 80000*8 waves exactly
    edge_scatter<<<80000, 256, 0, stream>>>(x, ei, ea, h0);

    // K3: h1 = relu(h0 @ W1 + b1)   100000 rows = 625*10*16 exactly
    gemm128<<<GEMM_BLOCKS, 256, lds_bytes, stream>>>(h0, W1, b1, h1, 1);

    // K4: out = h1 @ W2 + b2
    gemm128<<<GEMM_BLOCKS, 256, lds_bytes, stream>>>(h1, W2, b2, out, 0);
}